// ImpSentenceModel_60550448939113
// MI455X (gfx1250) — compile-verified
//
#include <hip/hip_runtime.h>
#include <hip/hip_bf16.h>

typedef __attribute__((ext_vector_type(2))) float v2f;
typedef __attribute__((ext_vector_type(8))) float v8f;

// Problem constants (from the reference setup)
constexpr int kB  = 32;
constexpr int kT  = 8192;
constexpr int kD  = 128;
constexpr int kML = 64;     // max_no_lines

constexpr int TOK_PER_WAVE  = 128;   // tokens handled by one wave32
constexpr int WAVES_PER_BLK = 8;     // 256 threads / block
constexpr int TOK_PER_BLOCK = TOK_PER_WAVE * WAVES_PER_BLK; // 1024

__global__ void zero_f32(float* __restrict__ p, int n) {
    int i = blockIdx.x * blockDim.x + threadIdx.x;
    if (i < n) p[i] = 0.0f;
}

// Segment-sum as one-hot WMMA:
//   out[b, s, :] += sum_{t : seg[b,t]==s} table[tok[b,t], :]
// A (16x4 f32)  = one-hot membership of 4 tokens against a 16-segment window
// B (4x16 f32)  = 4 embedding rows x 16 dims (one of 8 N-subtiles of D=128)
// C (16x16 f32) = 16 segments x 16 dims accumulator, exact f32 math.
__global__ void __launch_bounds__(256)
seg_embed_wmma(const int* __restrict__ tok,
               const int* __restrict__ seg,
               const float* __restrict__ table,
               float* __restrict__ out)
{
    __shared__ int s_seg[TOK_PER_BLOCK];

    const int tid  = threadIdx.x;
    const int lane = tid & 31;
    const int wave = tid >> 5;

    const int base = blockIdx.x * TOK_PER_BLOCK;   // flattened [B*T] offset
    const int b    = base / kT;                    // kT % TOK_PER_BLOCK == 0

    // Stage this block's segment ids into LDS (each wave reads only its slice).
    for (int i = tid; i < TOK_PER_BLOCK; i += 256)
        s_seg[i] = seg[base + i];
    __syncthreads();

    const int wbase = wave * TOK_PER_WAVE;         // LDS-local token base
    const int gbase = base + wbase;                // global token base
    const int seg_first = s_seg[wbase];
    const int seg_last  = s_seg[wbase + TOK_PER_WAVE - 1];

    const int mlane = lane & 15;   // M row (segment within window) / N col
    const int khalf = lane >> 4;   // selects K pair {0,1} vs {2,3}
    const int k0    = khalf * 2;

    // Sorted segments => contiguous window range; usually a single window.
    for (int w0 = (seg_first >> 4) << 4; w0 <= seg_last; w0 += 16) {
        v8f acc[8] = {};           // 16 segs x 128 dims, f32
        bool any = false;

        for (int kt = 0; kt < TOK_PER_WAVE; kt += 4) {
            // Wave-uniform intersection test (sorted within the 4-token tile).
            const int slo = s_seg[wbase + kt];
            const int shi = s_seg[wbase + kt + 3];
            if (shi < w0 || slo >= w0 + 16) continue;   // uniform branch
            any = true;

            const int sg0 = s_seg[wbase + kt + k0];
            const int sg1 = s_seg[wbase + kt + k0 + 1];

            // A: one-hot (exact in f32). Lane l<16: K=0/1, lane>=16: K=2/3.
            v2f a;
            a.x = (sg0 - w0 == mlane) ? 1.0f : 0.0f;
            a.y = (sg1 - w0 == mlane) ? 1.0f : 0.0f;

            // Embedding rows for this lane's two K slots (L2-resident table).
            const float* r0 = table + (size_t)tok[gbase + kt + k0]     * kD;
            const float* r1 = table + (size_t)tok[gbase + kt + k0 + 1] * kD;

            #pragma unroll
            for (int nt = 0; nt < 8; ++nt) {
                v2f bb;
                bb.x = r0[nt * 16 + mlane];
                bb.y = r1[nt * 16 + mlane];
                acc[nt] = __builtin_amdgcn_wmma_f32_16x16x4_f32(
                    /*neg_a=*/false, a, /*neg_b=*/false, bb,
                    /*c_mod=*/(short)0, acc[nt],
                    /*reuse_a=*/false, /*reuse_b=*/false);
            }
        }

        if (!any) continue;        // uniform: window had no tokens

        // Flush accumulator: C VGPR e holds M = e (+8 for lanes>=16), N = mlane.
        float* ob = out + ((size_t)b * kML + w0) * kD;
        #pragma unroll
        for (int nt = 0; nt < 8; ++nt) {
            #pragma unroll
            for (int e = 0; e < 8; ++e) {
                const int row = e + khalf * 8;     // segment within window
                atomicAdd(ob + (size_t)row * kD + nt * 16 + mlane, acc[nt][e]);
            }
        }
    }
}

extern "C" void kernel_launch(void* const* d_in, const int* in_sizes, int n_in,
                              void* d_out, int out_size, void* d_ws, size_t ws_size,
                              hipStream_t stream) {
    const int*   tokens = (const int*)d_in[0];     // paragraph_variable [B,T]
    const int*   segs   = (const int*)d_in[1];     // segment_ids        [B,T]
    // d_in[2] = max_no_lines scalar (== kML), unused
    const float* table  = (const float*)d_in[3];   // embedding_table    [V,D]
    float*       out    = (float*)d_out;           // [B, ML, D] f32

    // Output must be zeroed every call (atomic accumulation).
    zero_f32<<<(out_size + 255) / 256, 256, 0, stream>>>(out, out_size);

    const int blocks = (kB * kT) / TOK_PER_BLOCK;  // 256 workgroups
    seg_embed_wmma<<<blocks, 256, 0, stream>>>(tokens, segs, table, out);
}